// SinkhornMatching_64914135711961
// MI455X (gfx1250) — compile-verified
//
#include <hip/hip_runtime.h>
#include <math.h>

typedef __attribute__((ext_vector_type(2))) float v2f;
typedef __attribute__((ext_vector_type(8))) float v8f;

#define BATCH   16
#define DIM     256
#define NM      1024
#define MATELEMS ((size_t)BATCH * NM * NM)          // 16M floats per matrix
#define NORM_C  (-7.62461899f)                       // -log(1024+1024)
#define INV_SQRT_D 0.0625f                           // 1/sqrt(256)
#define NEG_BIG (-3.402823466e38f)

// ---------------------------------------------------------------------------
// GEMM: S[b,n,m] = (1/16) * sum_k A[b,k,n] * B[b,k,m]; also writes ST = S^T.
// Wave tile: 32(n) x 64(m) = 2x4 WMMA f32 16x16x4 tiles.
// Block: 8 waves as 4(n) x 2(m) -> 128 x 128 block tile. Grid: 8 x 8 x 16.
// ---------------------------------------------------------------------------
__global__ __launch_bounds__(256) void gemm_wmma_f32(
    const float* __restrict__ A, const float* __restrict__ B,
    float* __restrict__ S, float* __restrict__ ST) {
  const int lane = threadIdx.x & 31;
  const int wave = threadIdx.x >> 5;
  const int wr = wave >> 1;                 // 0..3 (n direction)
  const int wc = wave & 1;                  // 0..1 (m direction)
  const int n0 = blockIdx.x * 128 + wr * 32;
  const int m0 = blockIdx.y * 128 + wc * 64;
  const int b  = blockIdx.z;

  const float* Ab = A + (size_t)b * (DIM * NM);
  const float* Bb = B + (size_t)b * (DIM * NM);

  const int M  = lane & 15;                 // row-in-tile for A / col for B frag
  const int Kb = (lane >> 4) << 1;          // 0 or 2: K pair handled by this half-wave

  v8f acc[2][4] = {};

  for (int k = 0; k < DIM; k += 4) {
    const float* arow = Ab + (size_t)(k + Kb) * NM;
    const float* brow = Bb + (size_t)(k + Kb) * NM;
    v2f af[2], bf[4];
#pragma unroll
    for (int i = 0; i < 2; ++i) {
      af[i].x = arow[n0 + 16 * i + M];          // K = Kb
      af[i].y = arow[NM + n0 + 16 * i + M];     // K = Kb+1
    }
#pragma unroll
    for (int j = 0; j < 4; ++j) {
      bf[j].x = brow[m0 + 16 * j + M];
      bf[j].y = brow[NM + m0 + 16 * j + M];
    }
#pragma unroll
    for (int i = 0; i < 2; ++i)
#pragma unroll
      for (int j = 0; j < 4; ++j)
        acc[i][j] = __builtin_amdgcn_wmma_f32_16x16x4_f32(
            false, af[i], false, bf[j], (short)0, acc[i][j], false, false);
  }

  // Epilogue: C/D layout -> lane gives N = lane&15, VGPR r gives M = r + 8*(lane>>4)
  const int N = lane & 15;
  const int half = lane >> 4;
  const size_t bo = (size_t)b * (NM * NM);
#pragma unroll
  for (int i = 0; i < 2; ++i)
#pragma unroll
    for (int j = 0; j < 4; ++j)
#pragma unroll
      for (int r = 0; r < 8; ++r) {
        const int gn = n0 + 16 * i + r + 8 * half;
        const int gm = m0 + 16 * j + N;
        const float v = acc[i][j][r] * INV_SQRT_D;
        S [bo + (size_t)gn * NM + gm] = v;
        ST[bo + (size_t)gm * NM + gn] = v;
      }
}

// ---------------------------------------------------------------------------
// Zero the u/v dual-potential vectors (4 x 16384 floats).
// ---------------------------------------------------------------------------
__global__ void init_uv(float* __restrict__ w) {
  const int t = blockIdx.x * 256 + threadIdx.x;   // 64 blocks x 256
#pragma unroll
  for (int e = 0; e < 4; ++e) w[t * 4 + e] = 0.0f;
}

// ---------------------------------------------------------------------------
// One Sinkhorn half-step for BOTH directions:
//   out[row] = NORM - logsumexp_j( Smat[row, j] + add[b, j] )
// Row-major, fully coalesced 128-bit loads. One 256-thread block per row.
// Blocks [0,16384) -> fwd triple, [16384,32768) -> rev triple.
// ---------------------------------------------------------------------------
__global__ __launch_bounds__(256) void lse_pass(
    const float* __restrict__ Sf, const float* __restrict__ addf, float* __restrict__ outf,
    const float* __restrict__ Sr, const float* __restrict__ addr_, float* __restrict__ outr) {
  __shared__ float red[8];
  int row = blockIdx.x;
  const float* Sm; const float* av; float* ov;
  if (row < BATCH * NM) { Sm = Sf; av = addf; ov = outf; }
  else { row -= BATCH * NM; Sm = Sr; av = addr_; ov = outr; }
  const int b = row >> 10;
  const int t = threadIdx.x;

  const float* srow = Sm + (size_t)row * NM;
  const float* arow = av + (b << 10);

  const float4 s4 = *(const float4*)(srow + (t << 2));
  const float4 a4 = *(const float4*)(arow + (t << 2));
  float x0 = s4.x + a4.x, x1 = s4.y + a4.y, x2 = s4.z + a4.z, x3 = s4.w + a4.w;

  float mx = fmaxf(fmaxf(x0, x1), fmaxf(x2, x3));
#pragma unroll
  for (int off = 16; off; off >>= 1) mx = fmaxf(mx, __shfl_xor(mx, off, 32));
  if ((t & 31) == 0) red[t >> 5] = mx;
  __syncthreads();
  mx = red[0];
#pragma unroll
  for (int i = 1; i < 8; ++i) mx = fmaxf(mx, red[i]);
  __syncthreads();

  float s = __expf(x0 - mx) + __expf(x1 - mx) + __expf(x2 - mx) + __expf(x3 - mx);
#pragma unroll
  for (int off = 16; off; off >>= 1) s += __shfl_xor(s, off, 32);
  if ((t & 31) == 0) red[t >> 5] = s;
  __syncthreads();
  if (t == 0) {
    float tot = red[0];
#pragma unroll
    for (int i = 1; i < 8; ++i) tot += red[i];
    ov[row] = NORM_C - (mx + __logf(tot));
  }
}

// ---------------------------------------------------------------------------
// In-place finalize: out = exp(S + u + v - NORM_C) for both halves of d_out.
// ---------------------------------------------------------------------------
__global__ __launch_bounds__(256) void finalize(
    float* __restrict__ out,
    const float* __restrict__ uf, const float* __restrict__ vf,
    const float* __restrict__ ur, const float* __restrict__ vr) {
  const size_t idx = ((size_t)blockIdx.x * 256 + threadIdx.x) << 2;  // 4 elems/thread
  const int which = (int)(idx >> 24);                 // 0 = fwd, 1 = rev
  const size_t local = idx & (MATELEMS - 1);
  const int b = (int)(local >> 20);
  const int n = (int)((local >> 10) & 1023);
  const int m = (int)(local & 1023);
  const float* uvec = which ? ur : uf;
  const float* vvec = which ? vr : vf;
  const float u = uvec[(b << 10) + n];
  const float4 v4 = *(const float4*)(vvec + (b << 10) + m);
  float4 z = *(float4*)(out + idx);
  z.x = __expf(z.x + u + v4.x - NORM_C);
  z.y = __expf(z.y + u + v4.y - NORM_C);
  z.z = __expf(z.z + u + v4.z - NORM_C);
  z.w = __expf(z.w + u + v4.w - NORM_C);
  *(float4*)(out + idx) = z;
}

extern "C" void kernel_launch(void* const* d_in, const int* in_sizes, int n_in,
                              void* d_out, int out_size, void* d_ws, size_t ws_size,
                              hipStream_t stream) {
  const float* A = (const float*)d_in[0];   // batch_ad [16,256,1024]
  const float* B = (const float*)d_in[1];   // batch_bd [16,256,1024]
  float* out = (float*)d_out;
  float* S  = out;                          // fwd scores, evolves into exp(Z_fwd)
  float* ST = out + MATELEMS;               // rev scores (= S^T), -> exp(Z_rev)
  float* uf = (float*)d_ws;                 // [16,1024] each
  float* vf = uf + BATCH * NM;
  float* ur = uf + 2 * BATCH * NM;
  float* vr = uf + 3 * BATCH * NM;

  init_uv<<<64, 256, 0, stream>>>(uf);
  gemm_wmma_f32<<<dim3(8, 8, BATCH), 256, 0, stream>>>(A, B, S, ST);

  for (int it = 0; it < 100; ++it) {
    // u-update: fwd rows of S (+v_f), rev rows of ST (+v_r)
    lse_pass<<<2 * BATCH * NM, 256, 0, stream>>>(S, vf, uf, ST, vr, ur);
    // v-update: columns of S == rows of ST (+u_f); rev: rows of S (+u_r)
    lse_pass<<<2 * BATCH * NM, 256, 0, stream>>>(ST, uf, vf, S, ur, vr);
  }

  finalize<<<(unsigned)(2 * MATELEMS / (256 * 4)), 256, 0, stream>>>(out, uf, vf, ur, vr);
}